// TransE_1056561954978
// MI455X (gfx1250) — compile-verified
//
#include <hip/hip_runtime.h>
#include <hip/hip_bf16.h>
#include <math.h>

// ---------------- problem constants (from reference) ----------------
#define NUM_RELS 1315
#define FEAT_DIM 768
#define RD 32
#define OD 32
#define NENT 65536
#define NNZ_CNT 1048576
#define KTOT (FEAT_DIM + 2 * RD)   // 832
#define EPSN 1e-12f

typedef float v2f __attribute__((ext_vector_type(2)));
typedef float v8f __attribute__((ext_vector_type(8)));

// D = A(16x4,f32) * B(4x16,f32) + C(16x16,f32), full fp32 WMMA path.
__device__ __forceinline__ v8f wmma4(v2f a, v2f b, v8f c) {
    return __builtin_amdgcn_wmma_f32_16x16x4_f32(
        /*neg_a=*/false, a, /*neg_b=*/false, b,
        /*c_mod=*/(short)0, c, /*reuse_a=*/false, /*reuse_b=*/false);
}

// ---------------- prep: fold W_in/W_out/b_in/b_out into W_ent ----------------
// Wc[n][k] for k<768          = W_ent[n][k]
// Wc[n][768+c]                = sum_j W_ent[n][768+j] * W_in[j][c]
// Wc[n][800+c]                = sum_j W_ent[n][800+j] * W_out[j][c]
// bc[n] = b_ent[n] + sum_j W_ent[n][768+j]*b_in[j] + W_ent[n][800+j]*b_out[j]
__global__ void prep_kernel(const float* __restrict__ W_ent, const float* __restrict__ b_ent,
                            const float* __restrict__ W_in,  const float* __restrict__ b_in,
                            const float* __restrict__ W_out, const float* __restrict__ b_out,
                            float* __restrict__ Wc, float* __restrict__ bc) {
    int idx = blockIdx.x * blockDim.x + threadIdx.x;
    if (idx < OD * KTOT) {
        int n = idx / KTOT, k = idx % KTOT;
        float v;
        if (k < FEAT_DIM) {
            v = W_ent[n * KTOT + k];
        } else if (k < FEAT_DIM + RD) {
            int c = k - FEAT_DIM;
            float s = 0.f;
            for (int j = 0; j < RD; ++j) s += W_ent[n * KTOT + FEAT_DIM + j] * W_in[j * RD + c];
            v = s;
        } else {
            int c = k - FEAT_DIM - RD;
            float s = 0.f;
            for (int j = 0; j < RD; ++j) s += W_ent[n * KTOT + FEAT_DIM + RD + j] * W_out[j * RD + c];
            v = s;
        }
        Wc[idx] = v;
    } else if (idx < OD * KTOT + OD) {
        int n = idx - OD * KTOT;
        float s = b_ent[n];
        for (int j = 0; j < RD; ++j) {
            s += W_ent[n * KTOT + FEAT_DIM + j] * b_in[j];
            s += W_ent[n * KTOT + FEAT_DIM + RD + j] * b_out[j];
        }
        bc[n] = s;
    }
}

// ---------------- sparse aggregation: one wave per nonzero, lane d = dim d ---
// acc[row][d] += val * rel_emb[col][d]   (accumulators live in L2: 8MB each)
__global__ void agg_kernel(const int* __restrict__ rows, const int* __restrict__ cols,
                           const float* __restrict__ vals, const float* __restrict__ rel_emb,
                           float* __restrict__ acc) {
    unsigned gid = blockIdx.x * blockDim.x + threadIdx.x;
    unsigned nz = gid >> 5;           // same for all 32 lanes of a wave -> scalarizable
    unsigned d  = gid & 31u;
    int r = rows[nz];
    int c = cols[nz];
    float v = vals[nz];
    atomicAdd(&acc[(unsigned long long)r * 32ull + d], v * rel_emb[(unsigned)c * 32u + d]);
}

// ---------------- entity embedding GEMM + row-normalize ---------------------
// One wave: 32 rows x 32 cols using 4 WMMA accumulators. Logical K = 832:
//   [0,768)   A = feat, [768,800) A = acc_in, [800,832) A = acc_out, B = Wc.
__global__ __launch_bounds__(256) void ent_gemm_kernel(
    const float* __restrict__ feat_h, const float* __restrict__ feat_p, const float* __restrict__ feat_n,
    const float* __restrict__ ws_acc, const float* __restrict__ Wc, const float* __restrict__ bc,
    float* __restrict__ ws_ent) {
    const int side = blockIdx.y;
    const float* feat   = (side == 0) ? feat_h : (side == 1) ? feat_p : feat_n;
    const float* accin  = ws_acc + (size_t)(2 * side + 0) * NENT * 32;
    const float* accout = ws_acc + (size_t)(2 * side + 1) * NENT * 32;
    float* ent = ws_ent + (size_t)side * NENT * 32;

    const int wave = threadIdx.x >> 5;
    const int lane = threadIdx.x & 31;
    const int l15  = lane & 15;
    const int half = lane >> 4;     // 0: K pair {0,1}, 1: K pair {2,3} (ISA A layout)
    const int koff = half * 2;
    const int m0 = (blockIdx.x * 8 + wave) * 32;

    v8f c00 = {}, c01 = {}, c10 = {}, c11 = {};

    const float* a0p = feat + (size_t)(m0 + l15) * FEAT_DIM + koff;
    const float* a1p = feat + (size_t)(m0 + 16 + l15) * FEAT_DIM + koff;
    const float* b0p = Wc + (size_t)l15 * KTOT + koff;         // cols 0..15
    const float* b1p = Wc + (size_t)(16 + l15) * KTOT + koff;  // cols 16..31

#pragma unroll 4
    for (int k = 0; k < FEAT_DIM; k += 4) {
        v2f a0 = *(const v2f*)(a0p + k);
        v2f a1 = *(const v2f*)(a1p + k);
        v2f b0 = *(const v2f*)(b0p + k);
        v2f b1 = *(const v2f*)(b1p + k);
        c00 = wmma4(a0, b0, c00);
        c01 = wmma4(a0, b1, c01);
        c10 = wmma4(a1, b0, c10);
        c11 = wmma4(a1, b1, c11);
    }
    {
        const float* ai0 = accin + (size_t)(m0 + l15) * 32 + koff;
        const float* ai1 = accin + (size_t)(m0 + 16 + l15) * 32 + koff;
#pragma unroll
        for (int k = 0; k < RD; k += 4) {
            v2f a0 = *(const v2f*)(ai0 + k);
            v2f a1 = *(const v2f*)(ai1 + k);
            v2f b0 = *(const v2f*)(b0p + FEAT_DIM + k);
            v2f b1 = *(const v2f*)(b1p + FEAT_DIM + k);
            c00 = wmma4(a0, b0, c00);
            c01 = wmma4(a0, b1, c01);
            c10 = wmma4(a1, b0, c10);
            c11 = wmma4(a1, b1, c11);
        }
    }
    {
        const float* ao0 = accout + (size_t)(m0 + l15) * 32 + koff;
        const float* ao1 = accout + (size_t)(m0 + 16 + l15) * 32 + koff;
#pragma unroll
        for (int k = 0; k < RD; k += 4) {
            v2f a0 = *(const v2f*)(ao0 + k);
            v2f a1 = *(const v2f*)(ao1 + k);
            v2f b0 = *(const v2f*)(b0p + FEAT_DIM + RD + k);
            v2f b1 = *(const v2f*)(b1p + FEAT_DIM + RD + k);
            c00 = wmma4(a0, b0, c00);
            c01 = wmma4(a0, b1, c01);
            c10 = wmma4(a1, b0, c10);
            c11 = wmma4(a1, b1, c11);
        }
    }

    // bias + F.normalize(dim=1) + store. C layout: vgpr v, lanes 0-15 -> M=v,
    // lanes 16-31 -> M=v+8, N = l15 (+16 for second N tile).
    const float bias0 = bc[l15];
    const float bias1 = bc[16 + l15];
#pragma unroll
    for (int v = 0; v < 8; ++v) {
        float x0 = c00[v] + bias0, x1 = c01[v] + bias1;
        float s = x0 * x0 + x1 * x1;
        s += __shfl_xor(s, 1); s += __shfl_xor(s, 2); s += __shfl_xor(s, 4); s += __shfl_xor(s, 8);
        float sc = 1.0f / fmaxf(sqrtf(s), EPSN);
        size_t row = (size_t)(m0 + v + 8 * half);
        ent[row * 32 + l15]      = x0 * sc;
        ent[row * 32 + 16 + l15] = x1 * sc;

        float y0 = c10[v] + bias0, y1 = c11[v] + bias1;
        float t = y0 * y0 + y1 * y1;
        t += __shfl_xor(t, 1); t += __shfl_xor(t, 2); t += __shfl_xor(t, 4); t += __shfl_xor(t, 8);
        float tc = 1.0f / fmaxf(sqrtf(t), EPSN);
        size_t row2 = (size_t)(m0 + 16 + v + 8 * half);
        ent[row2 * 32 + l15]      = y0 * tc;
        ent[row2 * 32 + 16 + l15] = y1 * tc;
    }
}

// ---------------- relation projection + TransE scores -----------------------
// r = normalize(rel_emb[eid] @ W_rel.T + b_rel); scores = ||h + r - t||_2
__global__ __launch_bounds__(256) void score_kernel(
    const int* __restrict__ eid, const float* __restrict__ rel_emb,
    const float* __restrict__ W_rel, const float* __restrict__ b_rel,
    const float* __restrict__ ws_ent, float* __restrict__ out) {
    const int wave = threadIdx.x >> 5;
    const int lane = threadIdx.x & 31;
    const int l15  = lane & 15;
    const int half = lane >> 4;
    const int koff = half * 2;
    const int m0 = (blockIdx.x * 8 + wave) * 16;

    v8f c0 = {}, c1 = {};
    int e = eid[m0 + l15];  // gathered A rows (same rows in both halves, different K)
    const float* ap  = rel_emb + (size_t)e * RD + koff;
    const float* b0p = W_rel + (size_t)l15 * RD + koff;
    const float* b1p = W_rel + (size_t)(16 + l15) * RD + koff;
#pragma unroll
    for (int k = 0; k < RD; k += 4) {
        v2f a  = *(const v2f*)(ap + k);
        v2f b0 = *(const v2f*)(b0p + k);
        v2f b1 = *(const v2f*)(b1p + k);
        c0 = wmma4(a, b0, c0);
        c1 = wmma4(a, b1, c1);
    }

    const float* hE = ws_ent;
    const float* pE = ws_ent + (size_t)NENT * 32;
    const float* nE = ws_ent + 2ull * NENT * 32;
    const float bias0 = b_rel[l15];
    const float bias1 = b_rel[16 + l15];
#pragma unroll
    for (int v = 0; v < 8; ++v) {
        float r0 = c0[v] + bias0, r1 = c1[v] + bias1;
        float s = r0 * r0 + r1 * r1;
        s += __shfl_xor(s, 1); s += __shfl_xor(s, 2); s += __shfl_xor(s, 4); s += __shfl_xor(s, 8);
        float sc = 1.0f / fmaxf(sqrtf(s), EPSN);
        r0 *= sc; r1 *= sc;

        size_t row = (size_t)(m0 + v + 8 * half);
        float h0 = hE[row * 32 + l15], h1 = hE[row * 32 + 16 + l15];
        float p0 = pE[row * 32 + l15], p1 = pE[row * 32 + 16 + l15];
        float q0 = nE[row * 32 + l15], q1 = nE[row * 32 + 16 + l15];

        float dp0 = h0 + r0 - p0, dp1 = h1 + r1 - p1;
        float dn0 = h0 + r0 - q0, dn1 = h1 + r1 - q1;
        float sp = dp0 * dp0 + dp1 * dp1;
        sp += __shfl_xor(sp, 1); sp += __shfl_xor(sp, 2); sp += __shfl_xor(sp, 4); sp += __shfl_xor(sp, 8);
        float sn = dn0 * dn0 + dn1 * dn1;
        sn += __shfl_xor(sn, 1); sn += __shfl_xor(sn, 2); sn += __shfl_xor(sn, 4); sn += __shfl_xor(sn, 8);
        if (l15 == v) {
            out[row]        = sqrtf(sp);   // pos_score
            out[NENT + row] = sqrtf(sn);   // neg_score
        }
    }
}

// ---------------- launch --------------------------------------------------
extern "C" void kernel_launch(void* const* d_in, const int* in_sizes, int n_in,
                              void* d_out, int out_size, void* d_ws, size_t ws_size,
                              hipStream_t stream) {
    (void)in_sizes; (void)n_in; (void)out_size; (void)ws_size;

    // input order per setup_inputs()
    const int*   h_in_rows   = (const int*)  d_in[0];
    const int*   h_in_cols   = (const int*)  d_in[1];
    const float* h_in_vals   = (const float*)d_in[2];
    const int*   h_out_rows  = (const int*)  d_in[3];
    const int*   h_out_cols  = (const int*)  d_in[4];
    const float* h_out_vals  = (const float*)d_in[5];
    const float* h_feat      = (const float*)d_in[6];
    const int*   p_in_rows   = (const int*)  d_in[7];
    const int*   p_in_cols   = (const int*)  d_in[8];
    const float* p_in_vals   = (const float*)d_in[9];
    const int*   p_out_rows  = (const int*)  d_in[10];
    const int*   p_out_cols  = (const int*)  d_in[11];
    const float* p_out_vals  = (const float*)d_in[12];
    const float* p_feat      = (const float*)d_in[13];
    const int*   n_in_rows   = (const int*)  d_in[14];
    const int*   n_in_cols   = (const int*)  d_in[15];
    const float* n_in_vals   = (const float*)d_in[16];
    const int*   n_out_rows  = (const int*)  d_in[17];
    const int*   n_out_cols  = (const int*)  d_in[18];
    const float* n_out_vals  = (const float*)d_in[19];
    const float* n_feat      = (const float*)d_in[20];
    const int*   eid         = (const int*)  d_in[21];
    const float* rel_emb     = (const float*)d_in[22];
    const float* W_in        = (const float*)d_in[23];
    const float* b_in        = (const float*)d_in[24];
    const float* W_out       = (const float*)d_in[25];
    const float* b_out       = (const float*)d_in[26];
    const float* W_ent       = (const float*)d_in[27];
    const float* b_ent       = (const float*)d_in[28];
    const float* W_rel       = (const float*)d_in[29];
    const float* b_rel       = (const float*)d_in[30];

    // workspace layout (floats): [acc: 6*B*32][ent: 3*B*32][Wc: 32*832][bc: 32]  ~= 72 MB
    float* ws  = (float*)d_ws;
    float* acc = ws;
    float* ent = ws + 6ull * NENT * 32;
    float* Wc  = ws + 9ull * NENT * 32;
    float* bc  = Wc + (size_t)OD * KTOT;

    hipMemsetAsync(acc, 0, 6ull * NENT * 32 * sizeof(float), stream);

    prep_kernel<<<(OD * KTOT + OD + 255) / 256, 256, 0, stream>>>(
        W_ent, b_ent, W_in, b_in, W_out, b_out, Wc, bc);

    const unsigned agg_blocks = (unsigned)((unsigned long long)NNZ_CNT * 32ull / 256ull);  // 131072
    agg_kernel<<<agg_blocks, 256, 0, stream>>>(h_in_rows,  h_in_cols,  h_in_vals,  rel_emb, acc + 0ull * NENT * 32);
    agg_kernel<<<agg_blocks, 256, 0, stream>>>(h_out_rows, h_out_cols, h_out_vals, rel_emb, acc + 1ull * NENT * 32);
    agg_kernel<<<agg_blocks, 256, 0, stream>>>(p_in_rows,  p_in_cols,  p_in_vals,  rel_emb, acc + 2ull * NENT * 32);
    agg_kernel<<<agg_blocks, 256, 0, stream>>>(p_out_rows, p_out_cols, p_out_vals, rel_emb, acc + 3ull * NENT * 32);
    agg_kernel<<<agg_blocks, 256, 0, stream>>>(n_in_rows,  n_in_cols,  n_in_vals,  rel_emb, acc + 4ull * NENT * 32);
    agg_kernel<<<agg_blocks, 256, 0, stream>>>(n_out_rows, n_out_cols, n_out_vals, rel_emb, acc + 5ull * NENT * 32);

    // 32 rows per wave, 8 waves per block -> 256 blocks per side, 3 sides on y
    ent_gemm_kernel<<<dim3(NENT / (32 * 8), 3), 256, 0, stream>>>(
        h_feat, p_feat, n_feat, acc, Wc, bc, ent);

    // 16 rows per wave, 8 waves per block -> 512 blocks
    score_kernel<<<NENT / (16 * 8), 256, 0, stream>>>(
        eid, rel_emb, W_rel, b_rel, ent, (float*)d_out);
}